// RnCLoss_rot_sym_79147657331290
// MI455X (gfx1250) — compile-verified
//
#include <hip/hip_runtime.h>
#include <hip/hip_bf16.h>
#include <math.h>

typedef __attribute__((ext_vector_type(2))) float v2f;
typedef __attribute__((ext_vector_type(8))) float v8f;

#define TEMPERATURE   2.0f
#define INV_TEMP      0.5f
#define SL1_BETA      1.0f
#define EPS_F         1e-7f
#define MAX_N         512

// ---------------------------------------------------------------------------
// Kernel 1: per-row squared norms  sq[i] = sum_d F[i,d]^2   (one wave32 / row)
// ---------------------------------------------------------------------------
__global__ __launch_bounds__(256) void row_norms_kernel(
    const float* __restrict__ F, float* __restrict__ sq, int n, int feat) {
  const int wave = (blockIdx.x * blockDim.x + threadIdx.x) >> 5;
  const int lane = threadIdx.x & 31;
  if (wave >= n) return;
  const float* row = F + (size_t)wave * feat;
  float s = 0.0f;
  for (int d = lane; d < feat; d += 32) {
    float x = row[d];
    s = fmaf(x, x, s);
  }
  #pragma unroll
  for (int off = 16; off > 0; off >>= 1) s += __shfl_xor(s, off, 32);
  if (lane == 0) sq[wave] = s;
}

// ---------------------------------------------------------------------------
// Kernel 2: rot_diff[i,j] = mean_d smoothL1(rot[i,d]-rot[j,d])   (block per i)
// ---------------------------------------------------------------------------
__global__ __launch_bounds__(256) void rot_diff_kernel(
    const float* __restrict__ rot, float* __restrict__ diffm, int n, int rdim) {
  const int i = blockIdx.x;
  float ri[16];
  for (int d = 0; d < rdim; ++d) ri[d] = rot[(size_t)i * rdim + d];
  const float inv_r = 1.0f / (float)rdim;
  for (int j = threadIdx.x; j < n; j += blockDim.x) {
    float s = 0.0f;
    for (int d = 0; d < rdim; ++d) {
      float dv = ri[d] - rot[(size_t)j * rdim + d];
      float ad = fabsf(dv);
      s += (ad < SL1_BETA) ? (0.5f * dv * dv / SL1_BETA) : (ad - 0.5f * SL1_BETA);
    }
    diffm[(size_t)i * n + j] = s * inv_r;
  }
}

// ---------------------------------------------------------------------------
// Kernel 3: WMMA fp32 GEMM  G = F*F^T  (one wave per 16x16 tile), fused
// epilogue: logits = -sqrt(max(sq_i+sq_j-2G,0))/T ; exp = exp(logits).
// Row max is exactly 0 (diagonal), so max-subtraction is an exact no-op.
// A-frag (16x4 f32): lanes 0-15 -> {K,K+1} of row m0+L; lanes 16-31 -> {K+2,K+3}.
// B-frag of F*F^T has the identical pattern with column base n0.
// C/D (16x16 f32): VGPR v holds M=v (lanes 0-15) / M=v+8 (lanes 16-31), N=lane%16.
// ---------------------------------------------------------------------------
__global__ __launch_bounds__(32) void gemm_logits_kernel(
    const float* __restrict__ F, const float* __restrict__ sq,
    float* __restrict__ logits, float* __restrict__ expv, int n, int feat) {
  const int m0   = blockIdx.x * 16;
  const int n0   = blockIdx.y * 16;
  const int lane = threadIdx.x;       // 0..31, wave32
  const int r    = lane & 15;
  const int khl  = (lane >> 4) * 2;   // 0 for lanes 0-15, 2 for lanes 16-31

  const float* arow = F + (size_t)(m0 + r) * feat + khl;
  const float* brow = F + (size_t)(n0 + r) * feat + khl;

  v8f c = {};
  #pragma unroll 4
  for (int k = 0; k < feat; k += 4) {
    v2f a = *(const v2f*)(arow + k);   // 8B aligned: khl even, k % 4 == 0
    v2f b = *(const v2f*)(brow + k);
    c = __builtin_amdgcn_wmma_f32_16x16x4_f32(
        /*neg_a=*/false, a, /*neg_b=*/false, b,
        /*c_mod=*/(short)0, c, /*reuse_a=*/false, /*reuse_b=*/false);
  }

  const int mbase = m0 + ((lane >> 4) << 3);  // +8 for upper half-wave
  const int col   = n0 + r;
  const float sqc = sq[col];
  #pragma unroll
  for (int v = 0; v < 8; ++v) {
    const int row = mbase + v;
    float d2 = sq[row] + sqc - 2.0f * c[v];
    d2 = d2 > 0.0f ? d2 : 0.0f;
    float lg = -INV_TEMP * sqrtf(d2);
    logits[(size_t)row * n + col] = lg;
    expv  [(size_t)row * n + col] = __expf(lg);
  }
}

// ---------------------------------------------------------------------------
// Kernel 4: per row i:  denom[j] = sum_{k!=i} exp[i,k]*(diff[i,j] <= diff[i,k])
// partial[i] = sum_{j!=i} ( logits[i,j] - log(denom[j]+eps) )
// Rows cached in LDS; inner k-loop is a conflict-free LDS broadcast.
// ---------------------------------------------------------------------------
__global__ __launch_bounds__(256) void denom_loss_kernel(
    const float* __restrict__ logits, const float* __restrict__ expv,
    const float* __restrict__ diffm, float* __restrict__ partial, int n) {
  __shared__ float s_exp[MAX_N];
  __shared__ float s_diff[MAX_N];
  __shared__ float red[256];
  const int i = blockIdx.x;

  for (int k = threadIdx.x; k < n; k += blockDim.x) {
    s_exp[k]  = expv [(size_t)i * n + k];
    s_diff[k] = diffm[(size_t)i * n + k];
  }
  __syncthreads();
  if (threadIdx.x == 0) s_exp[i] = 0.0f;   // k == i contributes nothing
  __syncthreads();

  float acc = 0.0f;
  for (int j = threadIdx.x; j < n; j += blockDim.x) {
    if (j == i) continue;
    const float dj = s_diff[j];
    float den = 0.0f;
    #pragma unroll 4
    for (int k = 0; k < n; ++k) {
      den += (dj <= s_diff[k]) ? s_exp[k] : 0.0f;
    }
    acc += logits[(size_t)i * n + j] - __logf(den + EPS_F);
  }

  red[threadIdx.x] = acc;
  __syncthreads();
  for (int s = blockDim.x >> 1; s > 0; s >>= 1) {
    if (threadIdx.x < (unsigned)s) red[threadIdx.x] += red[threadIdx.x + s];
    __syncthreads();
  }
  if (threadIdx.x == 0) partial[i] = red[0];
}

// ---------------------------------------------------------------------------
// Kernel 5: deterministic fixed-order final reduction (no float atomics)
// ---------------------------------------------------------------------------
__global__ __launch_bounds__(256) void final_reduce_kernel(
    const float* __restrict__ partial, float* __restrict__ out, int n) {
  __shared__ float red[256];
  float s = 0.0f;
  for (int t = threadIdx.x; t < n; t += 256) s += partial[t];
  red[threadIdx.x] = s;
  __syncthreads();
  for (int st = 128; st > 0; st >>= 1) {
    if (threadIdx.x < (unsigned)st) red[threadIdx.x] += red[threadIdx.x + st];
    __syncthreads();
  }
  if (threadIdx.x == 0)
    out[0] = -red[0] / ((float)n * (float)(n - 1));
}

// ---------------------------------------------------------------------------
extern "C" void kernel_launch(void* const* d_in, const int* in_sizes, int n_in,
                              void* d_out, int out_size, void* d_ws, size_t ws_size,
                              hipStream_t stream) {
  const float* F   = (const float*)d_in[0];   // rot_feat [n, feat] f32
  const float* rot = (const float*)d_in[1];   // rot      [n, 9]    f32

  const int rdim = 9;
  const int n    = in_sizes[1] / rdim;        // 384
  const int feat = in_sizes[0] / n;           // 256
  if (n > MAX_N) return;

  float* ws      = (float*)d_ws;
  float* sq      = ws;                        // [n]
  float* logits  = sq + n;                    // [n*n]
  float* expv    = logits + (size_t)n * n;    // [n*n]
  float* diffm   = expv   + (size_t)n * n;    // [n*n]
  float* partial = diffm  + (size_t)n * n;    // [n]

  row_norms_kernel<<<(n * 32 + 255) / 256, 256, 0, stream>>>(F, sq, n, feat);
  rot_diff_kernel<<<n, 256, 0, stream>>>(rot, diffm, n, rdim);

  dim3 gtiles(n / 16, n / 16);
  gemm_logits_kernel<<<gtiles, 32, 0, stream>>>(F, sq, logits, expv, n, feat);

  denom_loss_kernel<<<n, 256, 0, stream>>>(logits, expv, diffm, partial, n);
  final_reduce_kernel<<<1, 256, 0, stream>>>(partial, (float*)d_out, n);
}